// DAGConvGNN_18743237280085
// MI455X (gfx1250) — compile-verified
//
#include <hip/hip_runtime.h>
#include <hip/hip_bf16.h>

#define DD 64
#define LF 12
#define LB 12

typedef __attribute__((ext_vector_type(2))) float v2f;
typedef __attribute__((ext_vector_type(8))) float v8f;

// ---------------- init: output buffer (-FLT_MAX | 0) and level counters ----------------
__global__ void init_kernel(float* __restrict__ out, int out_n,
                            int* __restrict__ counters, int ncnt) {
  int i = blockIdx.x * blockDim.x + threadIdx.x;
  int total = out_n + ncnt;
  for (; i < total; i += gridDim.x * blockDim.x) {
    if (i < out_n) {
      out[i] = ((i & 127) < DD) ? -3.402823466e+38f : 0.0f;  // [max_pool | sum_pool]
    } else {
      counters[i - out_n] = 0;
    }
  }
}

// ---------------- encoder: h = [nt, ninv] @ W_enc + b_enc ----------------
__global__ void encoder_kernel(const int* __restrict__ nt, const int* __restrict__ ninv,
                               const float* __restrict__ W_enc, const float* __restrict__ b_enc,
                               float* __restrict__ h, int N) {
  int i = blockIdx.x * blockDim.x + threadIdx.x;
  int total = N * 16;
  for (; i < total; i += gridDim.x * blockDim.x) {
    int v = i >> 4;
    int d0 = (i & 15) * 4;
    float a = (float)nt[v], b = (float)ninv[v];
    float4 r;
    r.x = a * W_enc[d0 + 0] + b * W_enc[DD + d0 + 0] + b_enc[d0 + 0];
    r.y = a * W_enc[d0 + 1] + b * W_enc[DD + d0 + 1] + b_enc[d0 + 1];
    r.z = a * W_enc[d0 + 2] + b * W_enc[DD + d0 + 2] + b_enc[d0 + 2];
    r.w = a * W_enc[d0 + 3] + b * W_enc[DD + d0 + 3] + b_enc[d0 + 3];
    *(float4*)(h + (size_t)v * DD + d0) = r;
  }
}

// ---------------- per level: compact level-l rows, zero agg rows + edge counts ----------------
__global__ void prep_kernel(const int* __restrict__ lvl, int l, int N,
                            int* __restrict__ rows, int* __restrict__ cnt,
                            float* __restrict__ agg, int* __restrict__ counter) {
  int v = blockIdx.x * blockDim.x + threadIdx.x;
  for (; v < N; v += gridDim.x * blockDim.x) {
    if (lvl[v] == l) {
      int idx = atomicAdd(counter, 1);
      rows[idx] = v;
      cnt[v] = 0;
      float4 z = {0.f, 0.f, 0.f, 0.f};
      float4* a = (float4*)(agg + (size_t)v * DD);
#pragma unroll
      for (int j = 0; j < 16; ++j) a[j] = z;
    }
  }
}

// ---------------- per level: agg[t] += h[g] over qualifying edges ----------------
__global__ void scatter_kernel(const int* __restrict__ gidx, const int* __restrict__ sidx,
                               const int* __restrict__ lvl, int l, int E,
                               const float* __restrict__ h, float* __restrict__ agg,
                               int* __restrict__ cnt) {
  int tid = blockIdx.x * blockDim.x + threadIdx.x;
  int total = E * 8;
  for (; tid < total; tid += gridDim.x * blockDim.x) {
    int e = tid >> 3, c = tid & 7;
    int t = sidx[e];
    if (lvl[t] != l) continue;
    int g = gidx[e];
    if (c == 0) atomicAdd(&cnt[t], 1);
    const float* hs = h + (size_t)g * DD + c * 8;
    float* ad = agg + (size_t)t * DD + c * 8;
#pragma unroll
    for (int j = 0; j < 8; ++j) atomicAdd(&ad[j], hs[j]);
  }
}

// ---------------- per level: h[rows] = agg[rows] @ W + cnt*b  (fp32 WMMA 16x16x4) ----------------
__global__ void __launch_bounds__(128) level_gemm_kernel(
    const float* __restrict__ agg, const int* __restrict__ rows,
    const int* __restrict__ cnt, const int* __restrict__ counter,
    const float* __restrict__ W, const float* __restrict__ bias,
    float* __restrict__ h) {
  __shared__ float ldsW[DD * DD];
  for (int i = threadIdx.x; i < DD * DD; i += blockDim.x) ldsW[i] = W[i];
  __syncthreads();

  const int lane = threadIdx.x & 31;
  const int rlane = lane & 15;
  const int hi = lane >> 4;                        // 0: lanes 0-15, 1: lanes 16-31
  const int waveGlobal = blockIdx.x * 4 + (threadIdx.x >> 5);
  const int numWaves = gridDim.x * 4;
  const int count = counter[0];
  const int numTiles = (count + 15) >> 4;

  for (int tile = waveGlobal; tile < numTiles; tile += numWaves) {
    const int base = tile << 4;
    int aIdx = base + rlane;
    if (aIdx >= count) aIdx = count - 1;           // clamp partial tile (stores masked later)
    const float* aRow = agg + (size_t)rows[aIdx] * DD;

    v8f acc0 = {}, acc1 = {}, acc2 = {}, acc3 = {};
#pragma unroll
    for (int k0 = 0; k0 < DD; k0 += 4) {
      const int ka = k0 + (hi << 1);
      // A 16x4: lanes 0-15 hold K=k0,k0+1; lanes 16-31 hold K=k0+2,k0+3 of row M=rlane
      v2f a = *(const v2f*)(aRow + ka);
      const float* w0 = ldsW + ka * DD;
      const float* w1 = ldsW + (ka + 1) * DD;
      // B 4x16: lanes 0-15 hold rows K=k0,k0+1 at col N=rlane; lanes 16-31 rows k0+2,k0+3
      v2f b0; b0.x = w0[rlane];       b0.y = w1[rlane];
      v2f b1; b1.x = w0[16 + rlane];  b1.y = w1[16 + rlane];
      v2f b2; b2.x = w0[32 + rlane];  b2.y = w1[32 + rlane];
      v2f b3; b3.x = w0[48 + rlane];  b3.y = w1[48 + rlane];
      acc0 = __builtin_amdgcn_wmma_f32_16x16x4_f32(false, a, false, b0, (short)0, acc0, false, false);
      acc1 = __builtin_amdgcn_wmma_f32_16x16x4_f32(false, a, false, b1, (short)0, acc1, false, false);
      acc2 = __builtin_amdgcn_wmma_f32_16x16x4_f32(false, a, false, b2, (short)0, acc2, false, false);
      acc3 = __builtin_amdgcn_wmma_f32_16x16x4_f32(false, a, false, b3, (short)0, acc3, false, false);
    }
    // C/D layout: vgpr r -> M=r (lanes 0-15, N=lane) / M=8+r (lanes 16-31, N=lane-16)
#pragma unroll
    for (int r = 0; r < 8; ++r) {
      int gm = base + (hi ? r + 8 : r);
      if (gm < count) {
        int node = rows[gm];
        float cf = (float)cnt[node];
        float* o = h + (size_t)node * DD;
        o[rlane]      = acc0[r] + cf * bias[rlane];
        o[16 + rlane] = acc1[r] + cf * bias[16 + rlane];
        o[32 + rlane] = acc2[r] + cf * bias[32 + rlane];
        o[48 + rlane] = acc3[r] + cf * bias[48 + rlane];
      }
    }
  }
}

// ---------------- readout: masked segment max/sum over batch ----------------
__device__ inline void atomicMaxFloat(float* addr, float value) {
  // valid when buffer is pre-initialized to a finite float; standard ordered-int trick
  if (value >= 0.0f) atomicMax((int*)addr, __float_as_int(value));
  else               atomicMin((unsigned int*)addr, (unsigned int)__float_as_int(value));
}

__global__ void readout_kernel(const float* __restrict__ h, const int* __restrict__ node_type,
                               const int* __restrict__ batch, float* __restrict__ out, int N) {
  int tid = blockIdx.x * blockDim.x + threadIdx.x;
  int total = N * DD;
  for (; tid < total; tid += gridDim.x * blockDim.x) {
    int v = tid >> 6, d = tid & 63;
    if (node_type[v] != 1) continue;
    int b = batch[v];
    float val = h[(size_t)v * DD + d];
    atomicMaxFloat(&out[b * 2 * DD + d], val);
    atomicAdd(&out[b * 2 * DD + DD + d], val);
  }
}

// ---------------- host launcher ----------------
extern "C" void kernel_launch(void* const* d_in, const int* in_sizes, int n_in,
                              void* d_out, int out_size, void* d_ws, size_t ws_size,
                              hipStream_t stream) {
  const int*   node_type = (const int*)d_in[0];
  const int*   ninv      = (const int*)d_in[1];
  const int*   eidx      = (const int*)d_in[2];
  const int*   flevel    = (const int*)d_in[3];
  const int*   blevel    = (const int*)d_in[4];
  const int*   batch     = (const int*)d_in[5];
  const float* W_enc     = (const float*)d_in[6];
  const float* b_enc     = (const float*)d_in[7];
  const float* W_f       = (const float*)d_in[8];
  const float* b_f       = (const float*)d_in[9];
  const float* W_b       = (const float*)d_in[10];
  const float* b_b       = (const float*)d_in[11];
  float* out = (float*)d_out;

  const int N = in_sizes[0];
  const int E = in_sizes[2] / 2;
  const int* esrc = eidx;
  const int* edst = eidx + E;

  char* ws = (char*)d_ws;
  size_t off = 0;
  float* h   = (float*)(ws + off); off += (size_t)N * DD * sizeof(float);
  float* agg = (float*)(ws + off); off += (size_t)N * DD * sizeof(float);
  int* cnt   = (int*)(ws + off);   off += (size_t)N * sizeof(int);
  int* rows  = (int*)(ws + off);   off += (size_t)N * sizeof(int);
  int* counters = (int*)(ws + off);

  const int nPrepBlk = (N + 255) / 256;
  const int nScatBlk = (E * 8 + 255) / 256;

  init_kernel<<<16, 256, 0, stream>>>(out, out_size, counters, 32);
  encoder_kernel<<<(N * 16 + 255) / 256, 256, 0, stream>>>(node_type, ninv, W_enc, b_enc, h, N);

  int slot = 0;
  for (int l = 1; l < LF; ++l, ++slot) {
    prep_kernel<<<nPrepBlk, 256, 0, stream>>>(flevel, l, N, rows, cnt, agg, counters + slot);
    scatter_kernel<<<nScatBlk, 256, 0, stream>>>(esrc, edst, flevel, l, E, h, agg, cnt);
    level_gemm_kernel<<<512, 128, 0, stream>>>(agg, rows, cnt, counters + slot, W_f, b_f, h);
  }
  for (int l = 1; l < LB; ++l, ++slot) {
    prep_kernel<<<nPrepBlk, 256, 0, stream>>>(blevel, l, N, rows, cnt, agg, counters + slot);
    scatter_kernel<<<nScatBlk, 256, 0, stream>>>(edst, esrc, blevel, l, E, h, agg, cnt);
    level_gemm_kernel<<<512, 128, 0, stream>>>(agg, rows, cnt, counters + slot, W_b, b_b, h);
  }

  readout_kernel<<<((size_t)N * DD + 255) / 256, 256, 0, stream>>>(h, node_type, batch, out, N);
}